// RFNO2d_72413148610912
// MI455X (gfx1250) — compile-verified
//
#include <hip/hip_runtime.h>
#include <math.h>

// ---------------------------------------------------------------------------
// RFNO2d forward for gfx1250 (MI455X), fp32 WMMA 16x16x4 path.
// Spectral conv implemented as truncated-DFT GEMMs (no FFT; 265=5*53 anyway).
// GEMM inner loop is branchless: M/N edges handled by address clamping
// (stores masked), K tails handled by zero-padded basis operands.
// ---------------------------------------------------------------------------

typedef float v2f __attribute__((ext_vector_type(2)));
typedef float v8f __attribute__((ext_vector_type(8)));

#define NP    265            // padded spatial size
#define NPIX  (265*265)      // 70225
#define NB    16             // batch
#define NC    32             // width
#define NM    13             // modes+1 (kept rfft cols)
#define NU    25             // 2*modes+1 (kept shifted rows)
#define NMODE (NU*NM)        // 325
#define NVC   (2*NM)         // 26 (re|im stacked)
#define ACTN  ((long)NB*NC*NPIX)

#define NP4   268            // ceil4(265): K-padded leading dims
#define NU4   28             // ceil4(25) and ceil4(26)

__device__ __forceinline__ float gelu_exact(float x) {
    return 0.5f * x * (1.0f + erff(x * 0.70710678118654752440f));
}

// ---------------------------------------------------------------------------
// Strided-batched GEMM, C = act(A*B + bias + res), row-major, f32 WMMA 16x16x4.
// One wave computes a 32x32 C tile; 4 v_wmma per K-step of 4.
// CONTRACT: loop reads k in [0, ceil4(K)); caller guarantees A or B is zero
// for k >= K (padded operand), and that reads up to 3 elements past logical
// ends stay inside the workspace (finite garbage * 0 = 0).
// Out-of-range rows/cols are clamped for loads; their results are never stored.
// ---------------------------------------------------------------------------
__global__ void gemm_wmma_f32(
    const float* __restrict__ A, const float* __restrict__ B, float* __restrict__ C,
    int M, int N, int K, int lda, int ldb, int ldc,
    long sA, long sB, long sC,
    const float* __restrict__ bias,
    const float* __restrict__ res, long sRes,
    int doGelu, int tilesM)
{
    const int tile = blockIdx.x;
    const int tm = (tile % tilesM) * 32;
    const int tn = (tile / tilesM) * 32;
    const long bz = blockIdx.y;
    A += bz * sA; B += bz * sB; C += bz * sC;
    const float* R = res ? (res + bz * sRes) : nullptr;

    const int lane = threadIdx.x & 31;
    const int half = lane >> 4;       // 0: K pair {0,1}, 1: K pair {2,3}
    const int l16  = lane & 15;

    v8f acc00 = {0.f,0.f,0.f,0.f,0.f,0.f,0.f,0.f};
    v8f acc01 = acc00, acc10 = acc00, acc11 = acc00;

    const int r0 = tm + l16;          // A rows, fragment 0
    const int r1 = tm + 16 + l16;     // A rows, fragment 1
    const int c0 = tn + l16;          // B cols, fragment 0
    const int c1 = tn + 16 + l16;     // B cols, fragment 1
    // clamp for loads; stores are masked below
    const int r0c = (r0 < M) ? r0 : (M - 1);
    const int r1c = (r1 < M) ? r1 : (M - 1);
    const int c0c = (c0 < N) ? c0 : (N - 1);
    const int c1c = (c1 < N) ? c1 : (N - 1);

    const float* pa0 = A + (long)r0c * lda + 2 * half;
    const float* pa1 = A + (long)r1c * lda + 2 * half;
    const float* pb0 = B + (long)(2 * half) * ldb + c0c;
    const float* pb1 = B + (long)(2 * half) * ldb + c1c;
    const long bstep = 4L * ldb;

    const int K4 = (K + 3) & ~3;
    #pragma unroll 2
    for (int k0 = 0; k0 < K4; k0 += 4) {
        v2f a0, a1, b0, b1;
        a0.x = pa0[0]; a0.y = pa0[1];
        a1.x = pa1[0]; a1.y = pa1[1];
        b0.x = pb0[0]; b0.y = pb0[ldb];
        b1.x = pb1[0]; b1.y = pb1[ldb];
        acc00 = __builtin_amdgcn_wmma_f32_16x16x4_f32(false, a0, false, b0, (short)0, acc00, false, false);
        acc01 = __builtin_amdgcn_wmma_f32_16x16x4_f32(false, a0, false, b1, (short)0, acc01, false, false);
        acc10 = __builtin_amdgcn_wmma_f32_16x16x4_f32(false, a1, false, b0, (short)0, acc10, false, false);
        acc11 = __builtin_amdgcn_wmma_f32_16x16x4_f32(false, a1, false, b1, (short)0, acc11, false, false);
        pa0 += 4; pa1 += 4; pb0 += bstep; pb1 += bstep;
    }

    // C/D layout: VGPR r -> row = r + 8*half, col = l16 (within 16x16 tile)
    #pragma unroll
    for (int r = 0; r < 8; ++r) {
        const int m0 = tm + r + 8 * half;
        const int m1 = m0 + 16;
        float v;
        if (m0 < M && c0 < N) {
            v = acc00[r];
            if (bias) v += bias[m0];
            if (R)    v += R[(long)m0 * ldc + c0];
            if (doGelu) v = gelu_exact(v);
            C[(long)m0 * ldc + c0] = v;
        }
        if (m0 < M && c1 < N) {
            v = acc01[r];
            if (bias) v += bias[m0];
            if (R)    v += R[(long)m0 * ldc + c1];
            if (doGelu) v = gelu_exact(v);
            C[(long)m0 * ldc + c1] = v;
        }
        if (m1 < M && c0 < N) {
            v = acc10[r];
            if (bias) v += bias[m1];
            if (R)    v += R[(long)m1 * ldc + c0];
            if (doGelu) v = gelu_exact(v);
            C[(long)m1 * ldc + c0] = v;
        }
        if (m1 < M && c1 < N) {
            v = acc11[r];
            if (bias) v += bias[m1];
            if (R)    v += R[(long)m1 * ldc + c1];
            if (doGelu) v = gelu_exact(v);
            C[(long)m1 * ldc + c1] = v;
        }
    }
}

// ---------------------------------------------------------------------------
// DFT basis construction (double trig; tiny cost). All operands K-padded:
// Ecol (268 x 26): rows 0..264 = [cos | -sin](2*pi*v*w/265); rows 265..267 = 0
// Ff   (50 x 268): cols 0..264 = cos rows 0..24 / sin rows 25..49; cols 265.. = 0
// Finv (530 x 28): cols 0..24 = cos rows 0..264 / sin rows 265..529; cols 25.. = 0
// Binv (28 x 265): rows 0..25 = [c_v*cos | -c_v*sin]/265^2, c_0=1 else 2; rows 26..27 = 0
// ---------------------------------------------------------------------------
__global__ void build_bases(float* __restrict__ Ecol, float* __restrict__ Ff,
                            float* __restrict__ Finv, float* __restrict__ Binv)
{
    int idx = blockIdx.x * blockDim.x + threadIdx.x;
    const double TP = 6.283185307179586476925286766559;
    const int nE = NP4 * NVC;        // 6968
    const int nF = 50 * NP4;         // 13400
    const int nI = (2 * NP) * NU4;   // 14840
    const int nBv = NU4 * NP;        // 7420
    if (idx < nE) {
        int w = idx / NVC, vc = idx % NVC, v = vc % NM;
        float val = 0.f;
        if (w < NP) {
            double th = TP * (double)(v * w) / (double)NP;
            val = (vc < NM) ? (float)cos(th) : (float)(-sin(th));
        }
        Ecol[idx] = val;
        return;
    }
    idx -= nE;
    if (idx < nF) {
        int row = idx / NP4, h = idx % NP4, u = row % NU;
        float val = 0.f;
        if (h < NP) {
            double ph = TP * (double)((u - 12) * h) / (double)NP;
            val = (row < NU) ? (float)cos(ph) : (float)sin(ph);
        }
        Ff[idx] = val;
        return;
    }
    idx -= nF;
    if (idx < nI) {
        int row = idx / NU4, u = idx % NU4, h = row % NP;
        float val = 0.f;
        if (u < NU) {
            double ph = TP * (double)((u - 12) * h) / (double)NP;
            val = (row < NP) ? (float)cos(ph) : (float)sin(ph);
        }
        Finv[idx] = val;
        return;
    }
    idx -= nI;
    if (idx < nBv) {
        int vc = idx / NP, w = idx % NP, v = vc % NM;
        float val = 0.f;
        if (vc < NVC) {
            double th = TP * (double)(v * w) / (double)NP;
            double sc = ((v == 0) ? 1.0 : 2.0) / (double)((long)NP * NP);
            val = (vc < NM) ? (float)(cos(th) * sc) : (float)(-sin(th) * sc);
        }
        Binv[idx] = val;
    }
}

// ---------------------------------------------------------------------------
// Build per-mode weight matrices Wt[l][mode=(u*13+v)][i][o] from W_LC/W_LR
// replicating build_weights(): W_LRc / W_UR / concat / rot90 logic.
// W_LC: (4,32,32,13,1)  W_LR: (4,32,32,12,12)
// ---------------------------------------------------------------------------
__global__ void build_Wt(const float* __restrict__ WLC, const float* __restrict__ WLR,
                         float* __restrict__ Wt)
{
    long idx = (long)blockIdx.x * blockDim.x + threadIdx.x;
    const long total = 4L * NMODE * 32 * 32;
    if (idx >= total) return;
    int l    = (int)(idx / ((long)NMODE * 1024));
    int rem  = (int)(idx % ((long)NMODE * 1024));
    int mode = rem / 1024;
    int io   = rem % 1024;
    int i = io / 32, o = io % 32;
    int u = mode / NM, v = mode % NM;
    long base13 = (((long)l * 32 + i) * 32 + o) * 13;
    long base12 = (((long)l * 32 + i) * 32 + o) * 144;
    float val;
    if (u >= 13) {                 // lower block = W_LRc
        int r = u - 13;
        val = (v == 0) ? WLC[base13 + (r + 1)] : WLR[base12 + r * 12 + (v - 1)];
    } else {                       // upper block = W_UR
        if (v == 0)       val = WLC[base13 + (12 - u)];
        else if (u == 12) val = WLC[base13 + v];                 // W_LRc[v-1][0]
        else              val = WLR[base12 + (v - 1) * 12 + (11 - u)];
    }
    Wt[idx] = val;
}

// lift (inC=1 conv1x1) + zero pad to 265x265, channel-first (b,c,h,w)
__global__ void lift_pad(const float* __restrict__ x, const float* __restrict__ pw,
                         const float* __restrict__ pb, float* __restrict__ out)
{
    long idx = (long)blockIdx.x * blockDim.x + threadIdx.x;
    if (idx >= ACTN) return;
    int p  = (int)(idx % NPIX);
    long bc = idx / NPIX;
    int c = (int)(bc % NC);
    int b = (int)(bc / NC);
    int h = p / NP, w = p % NP;
    float v = 0.f;
    if (h < 256 && w < 256)
        v = pw[c] * x[(long)b * 65536 + h * 256 + w] + pb[c];
    out[idx] = v;
}

// instance-norm stats per (b,c): mean and rsqrt(var+eps) over 265x265
__global__ void inorm_reduce(const float* __restrict__ x, float* __restrict__ stats)
{
    int bc = blockIdx.x;
    const float* p = x + (long)bc * NPIX;
    float s = 0.f, s2 = 0.f;
    for (int i = threadIdx.x; i < NPIX; i += 256) { float v = p[i]; s += v; s2 += v * v; }
    __shared__ float sh[256], sh2[256];
    sh[threadIdx.x] = s; sh2[threadIdx.x] = s2;
    __syncthreads();
    for (int st = 128; st > 0; st >>= 1) {
        if (threadIdx.x < st) { sh[threadIdx.x] += sh[threadIdx.x + st]; sh2[threadIdx.x] += sh2[threadIdx.x + st]; }
        __syncthreads();
    }
    if (threadIdx.x == 0) {
        float mu  = sh[0] / (float)NPIX;
        float var = sh2[0] / (float)NPIX - mu * mu;
        stats[2 * bc]     = mu;
        stats[2 * bc + 1] = rsqrtf(var + 1e-5f);
    }
}

__global__ void inorm_apply(const float* __restrict__ x, const float* __restrict__ stats,
                            float* __restrict__ y)
{
    long idx = (long)blockIdx.x * blockDim.x + threadIdx.x;
    if (idx >= ACTN) return;
    long bc = idx / NPIX;
    y[idx] = (x[idx] - stats[2 * bc]) * stats[2 * bc + 1];
}

// combine fwd-row GEMM halves into mode-major xs_m[mode][row(16 re +16 im)][i]
// xs_r = P[u,v] + Q[u,13+v] ; xs_i = P[u,13+v] - Q[u,v]   (PQ = [P;Q], 50x26)
__global__ void combine1(const float* __restrict__ PQ, float* __restrict__ xsm)
{
    long idx = (long)blockIdx.x * blockDim.x + threadIdx.x;
    const long total = (long)NMODE * 1024;
    if (idx >= total) return;
    int mode = (int)(idx / 1024);
    int r    = (int)((idx % 1024) / 32);
    int i    = (int)(idx % 32);
    int u = mode / NM, v = mode % NM;
    int b = r & 15;
    const float* pq = PQ + ((long)(b * NC + i)) * (50L * NVC);
    float val;
    if (r < 16) val = pq[u * NVC + v]        + pq[(NU + u) * NVC + NM + v];
    else        val = pq[u * NVC + NM + v]   - pq[(NU + u) * NVC + v];
    xsm[idx] = val;
}

// scatter ys_m[mode][b(re)/16+b(im)][o] into per-(b,o) matrices ys2[bo][u][re|im]
__global__ void combine2(const float* __restrict__ ysm, float* __restrict__ ys2)
{
    long idx = (long)blockIdx.x * blockDim.x + threadIdx.x;
    const long total = (long)NB * NC * NMODE;
    if (idx >= total) return;
    int b   = (int)(idx / ((long)NC * NMODE));
    int rem = (int)(idx % ((long)NC * NMODE));
    int o = rem / NMODE;
    int m = rem % NMODE;
    int u = m / NM, v = m % NM;
    long bo = (long)b * NC + o;
    const float* ym = ysm + (long)m * 1024;
    ys2[bo * (NU * NVC) + u * NVC + v]      = ym[b * 32 + o];
    ys2[bo * (NU * NVC) + u * NVC + NM + v] = ym[(16 + b) * 32 + o];
}

// combine inv-row GEMM halves: tmp_r = P[h,v]-Q[h,13+v]; tmp_i = P[h,13+v]+Q[h,v]
__global__ void combine3(const float* __restrict__ PQ2, float* __restrict__ tmp2)
{
    long idx = (long)blockIdx.x * blockDim.x + threadIdx.x;
    const long total = 512L * NP * NM;
    if (idx >= total) return;
    int bo  = (int)(idx / ((long)NP * NM));
    int rem = (int)(idx % ((long)NP * NM));
    int h = rem / NM, v = rem % NM;
    const float* pq = PQ2 + (long)bo * (2L * NP * NVC);
    float* tp = tmp2 + ((long)bo * NP + h) * NVC;
    tp[v]      = pq[h * NVC + v]      - pq[(NP + h) * NVC + NM + v];
    tp[NM + v] = pq[h * NVC + NM + v] + pq[(NP + h) * NVC + v];
}

// fused head: crop -> q1 (32->128) + gelu -> q2 (128->1); one thread per pixel
__global__ void q_fused(const float* __restrict__ x, const float* __restrict__ q1w,
                        const float* __restrict__ q1b, const float* __restrict__ q2w,
                        const float* __restrict__ q2b, float* __restrict__ out)
{
    __shared__ float w1[128 * 32];
    __shared__ float b1[128];
    __shared__ float w2[128];
    for (int t = threadIdx.x; t < 128 * 32; t += 256) w1[t] = q1w[t];
    if (threadIdx.x < 128) { b1[threadIdx.x] = q1b[threadIdx.x]; w2[threadIdx.x] = q2w[threadIdx.x]; }
    __syncthreads();

    long idx = (long)blockIdx.x * blockDim.x + threadIdx.x;
    const long total = (long)NB * 256 * 256;
    if (idx >= total) return;
    int b   = (int)(idx >> 16);
    int pix = (int)(idx & 65535);
    int h = pix >> 8, w = pix & 255;

    float xin[32];
    const float* xb = x + (long)b * NC * NPIX + (long)h * NP + w;
    #pragma unroll
    for (int i = 0; i < 32; ++i) xin[i] = xb[(long)i * NPIX];

    float acc = q2b[0];
    for (int j = 0; j < 128; ++j) {
        float a = b1[j];
        const float* wj = &w1[j * 32];
        #pragma unroll
        for (int i = 0; i < 32; ++i) a += wj[i] * xin[i];
        acc += w2[j] * gelu_exact(a);
    }
    out[idx] = acc;
}

// ---------------------------------------------------------------------------
extern "C" void kernel_launch(void* const* d_in, const int* in_sizes, int n_in,
                              void* d_out, int out_size, void* d_ws, size_t ws_size,
                              hipStream_t stream)
{
    (void)in_sizes; (void)n_in; (void)out_size; (void)ws_size;
    const float* x_in   = (const float*)d_in[0];
    const float* p_w    = (const float*)d_in[1];
    const float* p_b    = (const float*)d_in[2];
    const float* W_LC   = (const float*)d_in[3];
    const float* W_LR   = (const float*)d_in[4];
    const float* mlp1_w = (const float*)d_in[5];
    const float* mlp1_b = (const float*)d_in[6];
    const float* mlp2_w = (const float*)d_in[7];
    const float* mlp2_b = (const float*)d_in[8];
    const float* w_w    = (const float*)d_in[9];
    const float* w_b    = (const float*)d_in[10];
    const float* q1_w   = (const float*)d_in[11];
    const float* q1_b   = (const float*)d_in[12];
    const float* q2_w   = (const float*)d_in[13];
    const float* q2_b   = (const float*)d_in[14];

    float* ws = (float*)d_ws;
    float* actX = ws;                 // current activation  (b,c,h,w)
    float* actN = ws + ACTN;          // inorm(x) / hidden h1
    float* actS = ws + 2 * ACTN;      // spectral out s / x1
    float* actR = ws + 3 * ACTN;      // skip branch x2
    long off = 4 * ACTN;
    float* tbuf = ws + off; off += (long)NB * NC * NP * NVC;   // fwd col DFT out
    float* PQ   = ws + off; off += 512L * 50 * NVC;            // fwd row DFT out
    float* xsm  = ws + off; off += (long)NMODE * 1024;         // mode-major spectrum
    float* ysm  = ws + off; off += (long)NMODE * 1024;         // mode-mixed spectrum
    float* ys2  = ws + off; off += 512L * NU * NVC;            // per-(b,o) spectrum
    float* PQ2  = ws + off; off += 512L * (2 * NP) * NVC;      // inv row DFT out
    float* tmp2 = ws + off; off += (long)NB * NC * NP * NVC;   // inv row combined
    float* Ecol = ws + off; off += NP4 * NVC;                  // 268 x 26 (K-padded)
    float* Ff   = ws + off; off += 50 * NP4;                   // 50 x 268
    float* Finv = ws + off; off += (2 * NP) * NU4;             // 530 x 28
    float* Binv = ws + off; off += NU4 * NP;                   // 28 x 265
    float* Wt   = ws + off; off += 4L * NMODE * 1024;
    float* stats= ws + off; off += 1024;

    auto gemm = [&](const float* A, const float* B, float* C,
                    int M, int N, int K, int lda, int ldb, int ldc,
                    long sA, long sB, long sC, int batch,
                    const float* bias, const float* res, long sRes, int doGelu) {
        int tM = (M + 31) / 32, tN = (N + 31) / 32;
        dim3 grid((unsigned)(tM * tN), (unsigned)batch);
        gemm_wmma_f32<<<grid, dim3(32), 0, stream>>>(A, B, C, M, N, K, lda, ldb, ldc,
                                                     sA, sB, sC, bias, res, sRes, doGelu, tM);
    };

    // per-call precompute (deterministic, tiny)
    {
        const int nBase = NP4 * NVC + 50 * NP4 + (2 * NP) * NU4 + NU4 * NP;
        build_bases<<<(nBase + 255) / 256, 256, 0, stream>>>(Ecol, Ff, Finv, Binv);
        long n = 4L * NMODE * 1024;
        build_Wt<<<(unsigned)((n + 255) / 256), 256, 0, stream>>>(W_LC, W_LR, Wt);
    }
    lift_pad<<<(unsigned)((ACTN + 255) / 256), 256, 0, stream>>>(x_in, p_w, p_b, actX);

    const long ABS = (long)NC * NPIX;   // per-batch activation stride
    for (int i = 0; i < 4; ++i) {
        // n1 = inorm(x)
        inorm_reduce<<<512, 256, 0, stream>>>(actX, stats);
        inorm_apply<<<(unsigned)((ACTN + 255) / 256), 256, 0, stream>>>(actX, stats, actN);
        // fwd col DFT: t = n1(135680x265) * Ecol(268x26, rows>=265 zero)
        gemm(actN, Ecol, tbuf, NB * NC * NP, NVC, NP, NP, NVC, NVC, 0, 0, 0, 1,
             nullptr, nullptr, 0, 0);
        // fwd row DFT: PQ[bc](50x26) = Ff(50x268, cols>=265 zero) * t[bc](265x26)
        gemm(Ff, tbuf, PQ, 50, NVC, NP, NP4, NVC, NVC,
             0, (long)NP * NVC, 50L * NVC, 512, nullptr, nullptr, 0, 0);
        combine1<<<(unsigned)(((long)NMODE * 1024 + 255) / 256), 256, 0, stream>>>(PQ, xsm);
        // mode mixing: ys[mode](32x32) = xs[mode](32x32) * Wt[l][mode](32x32), batch 325
        gemm(xsm, Wt + (long)i * NMODE * 1024, ysm, 32, 32, 32, 32, 32, 32,
             1024, 1024, 1024, NMODE, nullptr, nullptr, 0, 0);
        combine2<<<(unsigned)(((long)NB * NC * NMODE + 255) / 256), 256, 0, stream>>>(ysm, ys2);
        // inv row DFT: PQ2[bo](530x26) = Finv(530x28, cols>=25 zero) * ys2[bo](25x26)
        gemm(Finv, ys2, PQ2, 2 * NP, NVC, NU, NU4, NVC, NVC,
             0, (long)NU * NVC, 2L * NP * NVC, 512, nullptr, nullptr, 0, 0);
        combine3<<<(unsigned)((512L * NP * NM + 255) / 256), 256, 0, stream>>>(PQ2, tmp2);
        // inv col DFT: s = tmp2(135680x26) * Binv(28x265, rows>=26 zero)
        gemm(tmp2, Binv, actS, NB * NC * NP, NP, NVC, NVC, NP, NP, 0, 0, 0, 1,
             nullptr, nullptr, 0, 0);
        // x1 = inorm(s)   (in place)
        inorm_reduce<<<512, 256, 0, stream>>>(actS, stats);
        inorm_apply<<<(unsigned)((ACTN + 255) / 256), 256, 0, stream>>>(actS, stats, actS);
        // h1 = gelu(mlp1 * x1 + b1)
        gemm(mlp1_w + i * 1024, actS, actN, NC, NPIX, NC, NC, NPIX, NPIX,
             0, ABS, ABS, NB, mlp1_b + i * NC, nullptr, 0, 1);
        // x2 = w * x + b (skip branch, reads actX before it is overwritten)
        gemm(w_w + i * 1024, actX, actR, NC, NPIX, NC, NC, NPIX, NPIX,
             0, ABS, ABS, NB, w_b + i * NC, nullptr, 0, 0);
        // x = act(mlp2 * h1 + b2 + x2)
        gemm(mlp2_w + i * 1024, actN, actX, NC, NPIX, NC, NC, NPIX, NPIX,
             0, ABS, ABS, NB, mlp2_b + i * NC, actR, ABS, (i < 3) ? 1 : 0);
    }

    // fused crop + q1 + gelu + q2
    q_fused<<<(unsigned)(((long)NB * 65536 + 255) / 256), 256, 0, stream>>>(
        actX, q1_w, q1_b, q2_w, q2_b, (float*)d_out);
}